// GCNConv_40252433498704
// MI455X (gfx1250) — compile-verified
//
#include <hip/hip_runtime.h>
#include <hip/hip_bf16.h>

// GCNConv: out = D^{-1/2} (I ∪ E) D^{-1/2} (X @ Theta)
// N=16384 nodes, E=524288 edges, D_IN=D_OUT=64, fp32.

#define N_NODES 16384
#define N_EDGES 524288
#define DFEAT   64
#define NBIT_WORDS (N_NODES * (N_NODES / 32))   // 2^23 words = 33.5 MB bitmap

typedef __attribute__((ext_vector_type(2))) float v2f;
typedef __attribute__((ext_vector_type(8))) float v8f;

// ---------------------------------------------------------------------------
// K1: x_theta = X @ Theta using V_WMMA_F32_16X16X4_F32.
// One wave per 16x16 output tile; K=64 consumed in 16 WMMA steps of 4.
// A (16x4 f32): lane = M + 16*(K/2), vgpr = K%2.  B (4x16): lane = N + 16*(K/2).
// D (16x16 f32): vgpr i -> M = i + 8*(lane>=16), N = lane&15.
// ---------------------------------------------------------------------------
__global__ void gcn_gemm_wmma(const float* __restrict__ x,
                              const float* __restrict__ theta,
                              float* __restrict__ xtheta) {
    const int lane = threadIdx.x & 31;
    const int wave = threadIdx.x >> 5;
    const int tile = blockIdx.x * 8 + wave;     // 4096 tiles = 1024 row x 4 col
    const int tm = tile >> 2;
    const int tn = tile & 3;
    const int r0 = tm * 16, c0 = tn * 16;
    const int mn = lane & 15;
    const int khalf = (lane >> 4) << 1;         // 0 or 2

    const float* __restrict__ xrow = x + (size_t)(r0 + mn) * DFEAT;
    v8f acc = {};
#pragma unroll
    for (int kk = 0; kk < DFEAT; kk += 4) {
        v2f a, b;
        a.x = xrow[kk + khalf + 0];
        a.y = xrow[kk + khalf + 1];
        b.x = theta[(size_t)(kk + khalf + 0) * DFEAT + c0 + mn];
        b.y = theta[(size_t)(kk + khalf + 1) * DFEAT + c0 + mn];
        acc = __builtin_amdgcn_wmma_f32_16x16x4_f32(
            /*neg_a=*/false, a, /*neg_b=*/false, b,
            /*c_mod=*/(short)0, acc, /*reuse_a=*/false, /*reuse_b=*/false);
    }
    const int row_base = r0 + ((lane >> 4) << 3);
    const int col = c0 + mn;
#pragma unroll
    for (int i = 0; i < 8; ++i)
        xtheta[(size_t)(row_base + i) * DFEAT + col] = acc[i];
}

// ---------------------------------------------------------------------------
// K2: init bitmap with diagonal (self-loops) pre-set, deg = 1, counter = 0.
// Word w holds bits [32w, 32w+32). Diagonal bit for row r is at r*16385;
// since 16385 > 32 at most one diagonal bit per word -> computed race-free.
// ---------------------------------------------------------------------------
__global__ void gcn_init(unsigned int* __restrict__ bitmap,
                         unsigned int* __restrict__ deg,
                         unsigned int* __restrict__ nuniq) {
    const unsigned int idx = blockIdx.x * blockDim.x + threadIdx.x;
    if (idx < (unsigned int)NBIT_WORDS) {
        const unsigned int w32 = idx << 5;
        const unsigned int r = (w32 + (N_NODES)) / (N_NODES + 1);  // ceil(w32/16385)
        const unsigned int pos = r * (N_NODES + 1);
        unsigned int val = 0u;
        if (r < N_NODES && pos >= w32 && pos < w32 + 32u)
            val = 1u << (pos - w32);
        bitmap[idx] = val;
    }
    if (idx < N_NODES) deg[idx] = 1u;   // self-loop contributes 1 to column sum
    if (idx == 0)      *nuniq = 0u;
}

// ---------------------------------------------------------------------------
// K3: dedup edges against the L2-resident bitmap. First setter of a bit
// bumps the column degree and appends the packed edge to the unique list.
// ---------------------------------------------------------------------------
__global__ void gcn_edges(const int* __restrict__ erow,
                          const int* __restrict__ ecol,
                          unsigned int* __restrict__ bitmap,
                          unsigned int* __restrict__ deg,
                          unsigned int* __restrict__ elist,
                          unsigned int* __restrict__ nuniq) {
    const unsigned int e = blockIdx.x * blockDim.x + threadIdx.x;
    if (e >= N_EDGES) return;
    const unsigned int r = (unsigned int)erow[e];
    const unsigned int c = (unsigned int)ecol[e];
    const unsigned int bit  = (r << 14) | c;          // r,c < 2^14 -> bit < 2^28
    const unsigned int word = bit >> 5;
    const unsigned int mask = 1u << (bit & 31u);
    const unsigned int old  = atomicOr(&bitmap[word], mask);
    if (!(old & mask)) {
        atomicAdd(&deg[c], 1u);                       // column-sum degree
        const unsigned int pos = atomicAdd(nuniq, 1u);
        elist[pos] = (r << 16) | c;
    }
}

// K4: dinv[i] = deg[i]^{-1/2}
__global__ void gcn_dinv(const unsigned int* __restrict__ deg,
                         float* __restrict__ dinv) {
    const int i = blockIdx.x * blockDim.x + threadIdx.x;
    if (i < N_NODES) dinv[i] = rsqrtf((float)deg[i]);
}

// K5: out[r,:] = dinv[r]^2 * xtheta[r,:]   (self-loop term, also inits out)
__global__ void gcn_selfloop(const float* __restrict__ xtheta,
                             const float* __restrict__ dinv,
                             float* __restrict__ out) {
    const int idx = blockIdx.x * blockDim.x + threadIdx.x;
    if (idx < N_NODES * DFEAT) {
        const int r = idx >> 6;
        const float s = dinv[r];
        out[idx] = s * s * xtheta[idx];
    }
}

// ---------------------------------------------------------------------------
// K6: scatter unique edges: out[r,:] += dinv[r]*dinv[c] * xtheta[c,:]
// One wave32 per edge; each lane handles features {lane, lane+32}.
// wid is wave-uniform: readfirstlane forces the edge word / counter onto the
// scalar path. unsafeAtomicAdd emits hardware GLOBAL_ATOMIC_ADD_F32 (no CAS
// loop); xtheta (4 MB) and out stay resident in the 192 MB L2.
// ---------------------------------------------------------------------------
__global__ void gcn_scatter(const unsigned int* __restrict__ elist,
                            const unsigned int* __restrict__ nuniq,
                            const float* __restrict__ xtheta,
                            const float* __restrict__ dinv,
                            float* __restrict__ out) {
    const unsigned int wid =
        __builtin_amdgcn_readfirstlane((blockIdx.x * blockDim.x + threadIdx.x) >> 5);
    const unsigned int lane = threadIdx.x & 31u;
    if (wid >= *nuniq) return;
    const unsigned int p = elist[wid];
    const unsigned int r = p >> 16;
    const unsigned int c = p & 0xFFFFu;
    const float coeff = dinv[r] * dinv[c];
    const float* __restrict__ src = xtheta + (size_t)c * DFEAT;
    float* __restrict__ dst = out + (size_t)r * DFEAT;
    unsafeAtomicAdd(&dst[lane],       coeff * src[lane]);
    unsafeAtomicAdd(&dst[lane + 32u], coeff * src[lane + 32u]);
}

extern "C" void kernel_launch(void* const* d_in, const int* in_sizes, int n_in,
                              void* d_out, int out_size, void* d_ws, size_t ws_size,
                              hipStream_t stream) {
    const float* x     = (const float*)d_in[0];          // [16384, 64]
    const int*   eidx  = (const int*)d_in[1];            // [2, 524288] flat
    const float* theta = (const float*)d_in[2];          // [64, 64]
    float* out = (float*)d_out;                          // [16384, 64]

    // Workspace layout (all 16B aligned), total ~40 MB:
    char* ws = (char*)d_ws;
    unsigned int* bitmap = (unsigned int*)(ws);                       // 33,554,432 B
    unsigned int* deg    = (unsigned int*)(ws + 33554432);            //     65,536 B
    float*        dinv   = (float*)       (ws + 33554432 + 65536);    //     65,536 B
    float*        xtheta = (float*)       (ws + 33554432 + 131072);   //  4,194,304 B
    unsigned int* elist  = (unsigned int*)(ws + 33554432 + 131072 + 4194304); // 2,097,152 B
    unsigned int* nuniq  = (unsigned int*)(ws + 33554432 + 131072 + 4194304 + 2097152);

    const int* erow = eidx;             // edge_index[0] = rows of A_hat
    const int* ecol = eidx + N_EDGES;   // edge_index[1] = cols of A_hat

    // K1: GEMM via fp32 WMMA — 4096 tiles, 8 waves/block
    gcn_gemm_wmma<<<512, 256, 0, stream>>>(x, theta, xtheta);
    // K2: bitmap (8,388,608 words) + deg + counter init
    gcn_init<<<NBIT_WORDS / 256, 256, 0, stream>>>(bitmap, deg, nuniq);
    // K3: dedup + degree + unique-edge compaction
    gcn_edges<<<N_EDGES / 256, 256, 0, stream>>>(erow, ecol, bitmap, deg, elist, nuniq);
    // K4: dinv = rsqrt(deg)
    gcn_dinv<<<N_NODES / 256, 256, 0, stream>>>(deg, dinv);
    // K5: out init with self-loop term
    gcn_selfloop<<<(N_NODES * DFEAT) / 256, 256, 0, stream>>>(xtheta, dinv, out);
    // K6: edge scatter, one wave per unique edge (guarded by *nuniq)
    gcn_scatter<<<(N_EDGES * 32) / 256, 256, 0, stream>>>(elist, nuniq, xtheta, dinv, out);
}